// GCNLayer_4999341932626
// MI455X (gfx1250) — compile-verified
//
#include <hip/hip_runtime.h>
#include <hip/hip_bf16.h>
#include <cstdint>

#define D_FEAT 128
#define TILE   256   // edges staged per LDS tile
#define K2_BLK 256   // 8 waves (wave32)

// ---------------------------------------------------------------------------
// CDNA5 helpers
// ---------------------------------------------------------------------------

// Low 32 bits of a generic pointer into LDS == LDS byte offset
// (aperture tag lives entirely in ADDR[63:32] on gfx1250).
__device__ __forceinline__ unsigned lds_off_u32(const void* p) {
  return (unsigned)(uintptr_t)p;
}

// Async DMA: global -> LDS, tracked on ASYNCcnt (gfx1250).
__device__ __forceinline__ void async_ld_b32(unsigned lds_byte_off, const void* gaddr) {
  asm volatile("global_load_async_to_lds_b32 %0, %1, off"
               :: "v"(lds_byte_off), "v"(gaddr)
               : "memory");
}

__device__ __forceinline__ void wait_async_0() {
  asm volatile("s_wait_asynccnt 0x0" ::: "memory");
}
__device__ __forceinline__ void wait_async_3() {
  asm volatile("s_wait_asynccnt 0x3" ::: "memory");
}

// Fire-and-forget f32 atomic adds (no return => STOREcnt only), saddr form.
// scope:SCOPE_DEV forces the RMW at the device-coherent point (L2 atomic
// units) -- required since many WGPs hit the same output rows.
__device__ __forceinline__ void atomic_add4_f32(float* base, unsigned byte_off,
                                                float x, float y, float z, float w) {
  asm volatile(
      "global_atomic_add_f32 %0, %1, %5 scope:SCOPE_DEV\n\t"
      "global_atomic_add_f32 %0, %2, %5 offset:4 scope:SCOPE_DEV\n\t"
      "global_atomic_add_f32 %0, %3, %5 offset:8 scope:SCOPE_DEV\n\t"
      "global_atomic_add_f32 %0, %4, %5 offset:12 scope:SCOPE_DEV"
      :: "v"(byte_off), "v"(x), "v"(y), "v"(z), "v"(w), "s"(base)
      : "memory");
}

__device__ __forceinline__ float gatef(float z) {
  // 2*sigmoid(z) - 1
  return 2.0f / (1.0f + __expf(-z)) - 1.0f;
}

// ---------------------------------------------------------------------------
// K2: per-edge gather * adj_val -> atomic scatter-add.
// Edge metadata double-buffered into LDS via async-to-LDS DMA (ASYNCcnt).
// One wave per edge instance: lane l owns features [4l, 4l+4).
// ---------------------------------------------------------------------------
template <bool PRE>
__global__ __launch_bounds__(K2_BLK) void k_spmm(const int* __restrict__ rowp,
                                                 const int* __restrict__ colp,
                                                 const float* __restrict__ valp,
                                                 const float* __restrict__ gated,
                                                 const float* __restrict__ embeds,
                                                 const float* __restrict__ zish,
                                                 float* __restrict__ out,
                                                 long long E, int ntiles) {
  __shared__ int   s_row[2][TILE];
  __shared__ int   s_col[2][TILE];
  __shared__ float s_val[2][TILE];

  const int t    = threadIdx.x;
  const int wave = t >> 5;
  const int lane = t & 31;
  const int stride = gridDim.x;
  const unsigned lane16 = (unsigned)lane << 4;  // lane * 16 bytes
  const int lane4 = lane << 2;                  // lane * 4 floats

  auto stage = [&](int tile, int buf) {
    long long e = (long long)tile * TILE + t;
    if (e >= E) e = E - 1;  // clamp: keep addresses valid, all lanes issue
    async_ld_b32(lds_off_u32(&s_row[buf][t]), rowp + e);
    async_ld_b32(lds_off_u32(&s_col[buf][t]), colp + e);
    async_ld_b32(lds_off_u32(&s_val[buf][t]), valp + e);
  };

  const int tile0 = blockIdx.x;
  if (tile0 >= ntiles) return;

  stage(tile0, 0);
  int buf = 0;
  for (int tile = tile0; tile < ntiles; tile += stride) {
    const int next = tile + stride;
    if (next < ntiles) {
      stage(next, buf ^ 1);  // prefetch next tile into the other buffer
      wait_async_3();        // our 3 oldest (current tile) have landed
    } else {
      wait_async_0();
    }
    __syncthreads();         // all waves' current-tile DMA complete

    const long long ebase = (long long)tile * TILE + wave * 32;
    const long long rem = E - ebase;
    const int cnt = rem >= 32 ? 32 : (rem > 0 ? (int)rem : 0);
    const int e0 = wave * 32;

    for (int i = 0; i < cnt; ++i) {
      const int   col = s_col[buf][e0 + i];
      const int   row = s_row[buf][e0 + i];
      const float a   = s_val[buf][e0 + i];

      float4 g;
      if (PRE) {
        g = *(const float4*)(gated + ((long long)col << 7) + lane4);
      } else {
        const float4 ev = *(const float4*)(embeds + ((long long)col << 7) + lane4);
        const float4 zv = *(const float4*)(zish   + ((long long)col << 7) + lane4);
        g.x = ev.x * gatef(zv.x);
        g.y = ev.y * gatef(zv.y);
        g.z = ev.z * gatef(zv.z);
        g.w = ev.w * gatef(zv.w);
      }

      // byte offset: row * 128 floats * 4B + lane * 16B  (fits in u32)
      const unsigned off = ((unsigned)row << 9) + lane16;
      atomic_add4_f32(out, off, a * g.x, a * g.y, a * g.z, a * g.w);
    }

    __syncthreads();  // reads of buf done before it is re-staged (t+2*stride)
    buf ^= 1;
  }
}

// Force early emission so the disasm snippet shows the async/atomic path.
template __global__ void k_spmm<true>(const int*, const int*, const float*,
                                      const float*, const float*, const float*,
                                      float*, long long, int);
template __global__ void k_spmm<false>(const int*, const int*, const float*,
                                       const float*, const float*, const float*,
                                       float*, long long, int);

// ---------------------------------------------------------------------------
// K1: gated = embeds * (2*sigmoid(z)-1); out = 0   (float4 vectorized)
// ---------------------------------------------------------------------------
__global__ __launch_bounds__(256) void k_gate_zero(const float* __restrict__ embeds,
                                                   const float* __restrict__ zish,
                                                   float* __restrict__ gated,
                                                   float* __restrict__ out,
                                                   long long total4, int write_gated) {
  long long i = (long long)blockIdx.x * blockDim.x + threadIdx.x;
  if (i >= total4) return;
  if (write_gated) {
    float4 e = ((const float4*)embeds)[i];
    float4 z = ((const float4*)zish)[i];
    float4 g;
    g.x = e.x * gatef(z.x);
    g.y = e.y * gatef(z.y);
    g.z = e.z * gatef(z.z);
    g.w = e.w * gatef(z.w);
    ((float4*)gated)[i] = g;
  }
  ((float4*)out)[i] = make_float4(0.f, 0.f, 0.f, 0.f);
}

// ---------------------------------------------------------------------------
// K3: leaky_relu in place; slope 0.5 < 1  =>  max(x, 0.5x)
// ---------------------------------------------------------------------------
__global__ __launch_bounds__(256) void k_leaky(float* __restrict__ out, long long total4) {
  long long i = (long long)blockIdx.x * blockDim.x + threadIdx.x;
  if (i >= total4) return;
  float4 v = ((float4*)out)[i];
  v.x = fmaxf(v.x, 0.5f * v.x);
  v.y = fmaxf(v.y, 0.5f * v.y);
  v.z = fmaxf(v.z, 0.5f * v.z);
  v.w = fmaxf(v.w, 0.5f * v.w);
  ((float4*)out)[i] = v;
}

// ---------------------------------------------------------------------------
extern "C" void kernel_launch(void* const* d_in, const int* in_sizes, int n_in,
                              void* d_out, int out_size, void* d_ws, size_t ws_size,
                              hipStream_t stream) {
  const int*   rowp = (const int*)d_in[0];
  const int*   colp = (const int*)d_in[1];
  const float* valp = (const float*)d_in[2];
  const float* emb  = (const float*)d_in[3];
  const float* zish = (const float*)d_in[4];
  float* out = (float*)d_out;

  const long long E      = in_sizes[0];
  const long long ND     = in_sizes[3];  // N * 128
  const long long total4 = ND / 4;

  const bool pre = (d_ws != nullptr) && (ws_size >= (size_t)ND * sizeof(float));
  float* gated = pre ? (float*)d_ws : nullptr;

  {
    const int blk = 256;
    const unsigned g = (unsigned)((total4 + blk - 1) / blk);
    k_gate_zero<<<g, blk, 0, stream>>>(emb, zish, gated, out, total4, pre ? 1 : 0);
  }
  {
    const int ntiles = (int)((E + TILE - 1) / TILE);
    const int grid = ntiles < 2048 ? ntiles : 2048;
    if (pre)
      k_spmm<true><<<grid, K2_BLK, 0, stream>>>(rowp, colp, valp, gated, emb, zish, out, E, ntiles);
    else
      k_spmm<false><<<grid, K2_BLK, 0, stream>>>(rowp, colp, valp, gated, emb, zish, out, E, ntiles);
  }
  {
    const int blk = 256;
    const unsigned g = (unsigned)((total4 + blk - 1) / blk);
    k_leaky<<<g, blk, 0, stream>>>(out, total4);
  }
}